// NN_84679575208444
// MI455X (gfx1250) — compile-verified
//
#include <hip/hip_runtime.h>
#include <math.h>

// ---------------------------------------------------------------------------
// Problem constants (from reference)
// ---------------------------------------------------------------------------
#define BB     16
#define NN     4096
#define KK     5
#define EPSV   1e-5f
#define SLOPE  0.2f
#define ROWS   (BB * NN)                 // 65536
#define CNTF   ((float)(ROWS * KK))      // 327680

typedef float v2f __attribute__((ext_vector_type(2)));
typedef float v8f __attribute__((ext_vector_type(8)));

// ---------------------------------------------------------------------------
// Workspace layout (float indices)
//   [0..8]    : stats  S0,S1,S2, M00,M01,M02,M11,M12,M22
//   [16..79]  : BN scale[64]
//   [80..143] : BN shift[64]
//   [256.. ]  : edges, ROWS * 16 floats (5 edges * 3, padded to 16)
// ---------------------------------------------------------------------------
#define WS_SCALE 16
#define WS_SHIFT 80
#define WS_EDGES 256

// ===========================================================================
// Kernel 0: zero the stat accumulators (must run every launch; ws not reset)
// ===========================================================================
__global__ void k_zero(float* __restrict__ ws) {
    if (threadIdx.x < 16) ws[threadIdx.x] = 0.0f;
}

// ===========================================================================
// Kernel 1: KNN (top-5 smallest distance, jax tie order) + edge vectors
//           + 9-scalar sufficient statistics for BatchNorm.
// grid: BB * (NN/256) = 256 blocks, 256 threads; one query point per thread.
// ===========================================================================
__global__ __launch_bounds__(256) void k_knn(const float* __restrict__ x,
                                             float* __restrict__ ws) {
    __shared__ float4 cand[256];

    const int tid = threadIdx.x;
    const int b   = blockIdx.x >> 4;
    const int n   = ((blockIdx.x & 15) << 8) + tid;
    const float* xb = x + (size_t)b * NN * 6;

    const float px = xb[n * 6 + 0];
    const float py = xb[n * 6 + 1];
    const float pz = xb[n * 6 + 2];
    const float sqn = px * px + py * py + pz * pz;

    // branchless sorted top-5 (ascending d); strict '<' keeps earlier index
    // first on ties, matching jax.lax.top_k(-d) ordering.
    float d0 = 3.4e38f, d1 = 3.4e38f, d2 = 3.4e38f, d3 = 3.4e38f, d4 = 3.4e38f;
    int   i0 = 0, i1 = 0, i2 = 0, i3 = 0, i4 = 0;

    for (int t = 0; t < NN / 256; ++t) {
        const int m0 = t << 8;
        {
            const float cx = xb[(m0 + tid) * 6 + 0];
            const float cy = xb[(m0 + tid) * 6 + 1];
            const float cz = xb[(m0 + tid) * 6 + 2];
            cand[tid] = make_float4(cx, cy, cz, cx * cx + cy * cy + cz * cz);
        }
        if (t + 1 < NN / 256)   // gfx1250 global_prefetch_b8 path
            __builtin_prefetch(&xb[(m0 + 256 + tid) * 6], 0, 1);
        __syncthreads();

        for (int j = 0; j < 256; ++j) {
            const float4 c = cand[j];
            const float dot = px * c.x + py * c.y + pz * c.z;
            const float dd  = sqn + c.w - 2.0f * dot;
            const int   m   = m0 + j;
            if (dd < d4) {
                if (dd < d3) {
                    d4 = d3; i4 = i3;
                    if (dd < d2) {
                        d3 = d2; i3 = i2;
                        if (dd < d1) {
                            d2 = d1; i2 = i1;
                            if (dd < d0) { d1 = d0; i1 = i0; d0 = dd; i0 = m; }
                            else         { d1 = dd; i1 = m; }
                        } else { d2 = dd; i2 = m; }
                    } else { d3 = dd; i3 = m; }
                } else { d4 = dd; i4 = m; }
            }
        }
        __syncthreads();
    }

    // edges + stat accumulation
    float s0 = 0.f, s1 = 0.f, s2 = 0.f;
    float m00 = 0.f, m01 = 0.f, m02 = 0.f, m11 = 0.f, m12 = 0.f, m22 = 0.f;
    float* er = ws + WS_EDGES + (size_t)(b * NN + n) * 16;
    const int idx[KK] = { i0, i1, i2, i3, i4 };
#pragma unroll
    for (int k = 0; k < KK; ++k) {
        const int m = idx[k];
        const float e0 = xb[m * 6 + 0] - px;
        const float e1 = xb[m * 6 + 1] - py;
        const float e2 = xb[m * 6 + 2] - pz;
        er[k * 3 + 0] = e0; er[k * 3 + 1] = e1; er[k * 3 + 2] = e2;
        s0 += e0; s1 += e1; s2 += e2;
        m00 += e0 * e0; m01 += e0 * e1; m02 += e0 * e2;
        m11 += e1 * e1; m12 += e1 * e2; m22 += e2 * e2;
    }

    float vals[9] = { s0, s1, s2, m00, m01, m02, m11, m12, m22 };
#pragma unroll
    for (int s = 0; s < 9; ++s) {
        float v = vals[s];
#pragma unroll
        for (int off = 16; off > 0; off >>= 1) v += __shfl_xor(v, off, 32);
        if ((tid & 31) == 0) atomicAdd(&ws[s], v);
    }
}

// ===========================================================================
// Kernel 2: fold BN(train) into per-channel affine using sufficient stats.
// 1 block, 64 threads.
// ===========================================================================
__global__ void k_bn(const float* __restrict__ conv_w,
                     const float* __restrict__ g,
                     const float* __restrict__ bta,
                     float* __restrict__ ws) {
    const int c = threadIdx.x;
    const float s0 = ws[0], s1 = ws[1], s2 = ws[2];
    const float m00 = ws[3], m01 = ws[4], m02 = ws[5];
    const float m11 = ws[6], m12 = ws[7], m22 = ws[8];
    const float w0 = conv_w[c], w1 = conv_w[64 + c], w2 = conv_w[128 + c];
    const float inv = 1.0f / CNTF;
    const float mean = (s0 * w0 + s1 * w1 + s2 * w2) * inv;
    const float eh2  = (m00 * w0 * w0 + m11 * w1 * w1 + m22 * w2 * w2 +
                        2.0f * (m01 * w0 * w1 + m02 * w0 * w2 + m12 * w1 * w2)) * inv;
    const float var  = eh2 - mean * mean;
    const float rs   = rsqrtf(var + EPSV);
    const float sc   = g[c] * rs;
    ws[WS_SCALE + c] = sc;
    ws[WS_SHIFT + c] = bta[c] - mean * sc;
}

// ===========================================================================
// Kernel 3: fused edge-conv/BN/leaky/max + 6-layer MLP on fp32 WMMA.
// Block = 64 threads (2 waves). Each wave owns 16 rows through all layers.
// Per-wave LDS ping-pong buffers, strides chosen so A-fragment reads hit
// 32 distinct banks (stride % 64 banks == 4-bank step per lane).
// ===========================================================================
#define ST0 132   // buffer0 stride (width <= 128)
#define ST1 260   // buffer1 stride (width <= 256)

// One GEMM layer: out[16 x Cout] = relu?(in[16 x Cin] @ W + bias)
// WMMA fragment layouts per CDNA5 ISA 7.12.2 (16x16x4 f32):
//   A: lanes 0-15 row=lane K={k,k+1}; lanes 16-31 row=lane-16 K={k+2,k+3}
//   B: column-major mirror of A (lane=col, same K split)
//   C/D: vgpr i -> row i (lanes<16) / row 8+i (lanes>=16); lane&15 = col
__device__ __forceinline__ void mlp_layer(const float* __restrict__ W,
                                          const float* __restrict__ bias,
                                          const float* __restrict__ in, int sIn,
                                          float* __restrict__ out, int sOut,
                                          int Cin, int Cout, int lane, bool relu) {
    const int col   = lane & 15;
    const int hi    = lane >> 4;       // 0 or 1
    const int koff  = hi * 2;
    const int mbase = hi * 8;
    for (int j = 0; j < Cout; j += 16) {
        const float bv = bias[j + col];
        v8f acc = { bv, bv, bv, bv, bv, bv, bv, bv };
        for (int k = 0; k < Cin; k += 4) {
            v2f a, bf;
            a.x = in[col * sIn + k + koff];
            a.y = in[col * sIn + k + koff + 1];
            const int kk = k + koff;
            bf.x = W[kk * Cout + j + col];
            bf.y = W[(kk + 1) * Cout + j + col];
            acc = __builtin_amdgcn_wmma_f32_16x16x4_f32(
                false, a, false, bf, (short)0, acc, false, false);
        }
#pragma unroll
        for (int i = 0; i < 8; ++i) {
            float v = acc[i];
            if (relu) v = fmaxf(v, 0.0f);
            out[(mbase + i) * sOut + j + col] = v;
        }
    }
}

__global__ __launch_bounds__(64) void k_mlp(
    const float* __restrict__ ws, const float* __restrict__ conv_w,
    const float* __restrict__ w1, const float* __restrict__ b1,
    const float* __restrict__ w2, const float* __restrict__ b2,
    const float* __restrict__ w3, const float* __restrict__ b3,
    const float* __restrict__ w4, const float* __restrict__ b4,
    const float* __restrict__ w5, const float* __restrict__ b5,
    const float* __restrict__ w6, const float* __restrict__ b6,
    float* __restrict__ out) {
    __shared__ float sb0[2][16 * ST0];   // 16.5 KB
    __shared__ float sb1[2][16 * ST1];   // 32.5 KB  (total ~49 KB / block)

    const int wave = threadIdx.x >> 5;
    const int lane = threadIdx.x & 31;
    const int row0 = blockIdx.x * 32 + wave * 16;
    float* buf0 = sb0[wave];
    float* buf1 = sb1[wave];

    // ---- stage 1: edge conv + BN affine + max_k + leaky -> x1 into buf0 ----
    {
        const int c0 = lane * 2, c1 = c0 + 1;
        const float wa0 = conv_w[c0], wa1 = conv_w[64 + c0], wa2 = conv_w[128 + c0];
        const float wb0 = conv_w[c1], wb1 = conv_w[64 + c1], wb2 = conv_w[128 + c1];
        const float sc0 = ws[WS_SCALE + c0], sh0 = ws[WS_SHIFT + c0];
        const float sc1 = ws[WS_SCALE + c1], sh1 = ws[WS_SHIFT + c1];
        for (int r = 0; r < 16; ++r) {
            const float* e = ws + WS_EDGES + (size_t)(row0 + r) * 16;
            float mx0 = -3.4e38f, mx1 = -3.4e38f;
#pragma unroll
            for (int k = 0; k < KK; ++k) {
                const float e0 = e[k * 3 + 0], e1 = e[k * 3 + 1], e2 = e[k * 3 + 2];
                const float h0 = e0 * wa0 + e1 * wa1 + e2 * wa2;
                const float h1 = e0 * wb0 + e1 * wb1 + e2 * wb2;
                mx0 = fmaxf(mx0, h0 * sc0 + sh0);
                mx1 = fmaxf(mx1, h1 * sc1 + sh1);
            }
            buf0[r * ST0 + c0] = mx0 > 0.0f ? mx0 : SLOPE * mx0;  // leaky(max)==max(leaky)
            buf0[r * ST0 + c1] = mx1 > 0.0f ? mx1 : SLOPE * mx1;
        }
    }
    __syncthreads();

    // ---- MLP layers on fp32 WMMA ----
    mlp_layer(w1, b1, buf0, ST0, buf1, ST1, 64, 64, lane, true);   __syncthreads();
    mlp_layer(w2, b2, buf1, ST1, buf0, ST0, 64, 128, lane, true);  __syncthreads();
    mlp_layer(w3, b3, buf0, ST0, buf1, ST1, 128, 256, lane, true); __syncthreads();
    mlp_layer(w4, b4, buf1, ST1, buf0, ST0, 256, 128, lane, true); __syncthreads();
    mlp_layer(w5, b5, buf0, ST0, buf1, ST1, 128, 64, lane, true);  __syncthreads();

    // ---- final layer 64 -> 13 (padded to 16 cols), straight to global ----
    {
        const int col = lane & 15;
        const int hi  = lane >> 4;
        const int koff = hi * 2;
        const int mbase = hi * 8;
        const bool valid = col < 13;
        const float bv = valid ? b6[col] : 0.0f;
        v8f acc = { bv, bv, bv, bv, bv, bv, bv, bv };
        for (int k = 0; k < 64; k += 4) {
            v2f a, bf;
            a.x = buf1[col * ST1 + k + koff];
            a.y = buf1[col * ST1 + k + koff + 1];
            const int kk = k + koff;
            bf.x = valid ? w6[kk * 13 + col] : 0.0f;
            bf.y = valid ? w6[(kk + 1) * 13 + col] : 0.0f;
            acc = __builtin_amdgcn_wmma_f32_16x16x4_f32(
                false, a, false, bf, (short)0, acc, false, false);
        }
        if (valid) {
#pragma unroll
            for (int i = 0; i < 8; ++i)
                out[(size_t)(row0 + mbase + i) * 13 + col] = acc[i];
        }
    }
}

// ===========================================================================
// Launch
// ===========================================================================
extern "C" void kernel_launch(void* const* d_in, const int* in_sizes, int n_in,
                              void* d_out, int out_size, void* d_ws, size_t ws_size,
                              hipStream_t stream) {
    const float* x      = (const float*)d_in[0];
    const float* conv_w = (const float*)d_in[1];
    const float* bn_g   = (const float*)d_in[2];
    const float* bn_b   = (const float*)d_in[3];
    const float* w1 = (const float*)d_in[4];  const float* b1 = (const float*)d_in[5];
    const float* w2 = (const float*)d_in[6];  const float* b2 = (const float*)d_in[7];
    const float* w3 = (const float*)d_in[8];  const float* b3 = (const float*)d_in[9];
    const float* w4 = (const float*)d_in[10]; const float* b4 = (const float*)d_in[11];
    const float* w5 = (const float*)d_in[12]; const float* b5 = (const float*)d_in[13];
    const float* w6 = (const float*)d_in[14]; const float* b6 = (const float*)d_in[15];
    float* ws  = (float*)d_ws;
    float* out = (float*)d_out;

    k_zero<<<1, 32, 0, stream>>>(ws);
    k_knn<<<BB * (NN / 256), 256, 0, stream>>>(x, ws);
    k_bn<<<1, 64, 0, stream>>>(conv_w, bn_g, bn_b, ws);
    k_mlp<<<ROWS / 32, 64, 0, stream>>>(ws, conv_w, w1, b1, w2, b2, w3, b3,
                                        w4, b4, w5, b5, w6, b6, out);
}